// CARAFENaive_5446018531572
// MI455X (gfx1250) — compile-verified
//
#include <hip/hip_runtime.h>

typedef __attribute__((ext_vector_type(2))) float v2f;
typedef __attribute__((ext_vector_type(8))) float v8f;

#define KTAP 5
#define RPAD 2
#define CCH  128
#define HH   128
#define WW   128
#define OHH  256
#define OWW  256
#define NCT  8                 // 8 channel tiles of 16 -> all 128 channels per wave
#define TILE_W 16              // low-res pixels per block (4 waves x 4 px)
#define LDS_COLS (TILE_W + 2*RPAD)            // 20
#define LDS_ROWS (KTAP)                       // 5
#define LDS_CH  (LDS_ROWS * LDS_COLS)         // 100 dwords/channel: bank-conflict-free
#define LDS_TOT (CCH * LDS_CH)                // 12800 floats = 50 KB
#define OST 66                 // output-repack channel stride (dwords): halves hit
                               // disjoint banks (8*66 mod 64 == 16)

__global__ __launch_bounds__(128)
void carafe_wmma_f32_kernel(const float* __restrict__ feats,
                            const float* __restrict__ masks,
                            float* __restrict__ out) {
    __shared__ float lfeat[LDS_TOT];          // reused for output repack (needs 8448)

    const int tid  = threadIdx.x;
    const int lane = tid & 31;
    const int wave = tid >> 5;
    const int half = lane >> 4;      // lanes 16-31 hold K rows 2,3
    const int mrow = lane & 15;      // A: channel row; B/C/D: column N
    const int p    = mrow >> 2;      // low-res pixel 0..3 within wave group
    const int s    = mrow & 3;       // subpixel
    const int s1   = s >> 1;
    const int s2   = s & 1;

    const int bid  = blockIdx.x;     // grid = n(2) * hi(128) * wgrp(8)
    const int wgrp = bid & 7;
    const int hi   = (bid >> 3) & 127;
    const int n    = bid >> 10;
    const int wblk = wgrp * TILE_W;

    // ---- stage zero-padded feature tile: 128ch x 5rows x 20cols -> LDS ----
    // Even-offset b64 pairs never straddle the pad/valid boundary (all even).
    const float* feat_n = feats + (size_t)n * CCH * HH * WW;
    for (int j = tid; j < LDS_TOT / 2; j += 128) {        // exactly 50 iters/thread
        const int c   = j / 50;
        const int rem = j - c * 50;
        const int r   = rem / 10;
        const int x   = (rem - r * 10) * 2;
        const int gr  = hi + r - RPAD;
        const int gxc = wblk + x - RPAD;                  // always even
        v2f v = {0.f, 0.f};
        if ((unsigned)gr < (unsigned)HH && (unsigned)gxc < (unsigned)WW)
            v = *(const v2f*)(feat_n + (size_t)c * (HH * WW) + (size_t)gr * WW + gxc);
        *(v2f*)(lfeat + 2 * j) = v;                       // 8B-aligned
    }
    __syncthreads();

    // ---- main loop: 25 taps x 8 channel tiles of v_wmma_f32_16x16x4_f32 ----
    const int oh  = 2 * hi + s1;
    const int ow  = 2 * (wblk + wave * 4 + p) + s2;

    // B gating: VGPR0 holds K=(half?2:0), VGPR1 holds K=(half?3:1);
    // column N=(p,s) has its single nonzero at K==p.
    const bool gbx = (p == (half ? 2 : 0));
    const bool gby = (p == (half ? 3 : 1));

    v8f acc[NCT];
    const v8f vzero = {0.f, 0.f, 0.f, 0.f, 0.f, 0.f, 0.f, 0.f};
#pragma unroll
    for (int t = 0; t < NCT; ++t) acc[t] = vzero;

    const float* mask_base = masks + (size_t)n * (KTAP * KTAP) * (OHH * OWW)
                                   + (size_t)oh * OWW + ow;
    const int xbase = wave * 4 + half * 2;    // this lane's two K columns (in LDS)

#pragma unroll
    for (int kk = 0; kk < KTAP * KTAP; ++kk) {
        const int di = kk / KTAP;
        const int dj = kk - di * KTAP;

        const float mval = mask_base[(size_t)kk * (OHH * OWW)];
        v2f b;
        b.x = gbx ? mval : 0.f;
        b.y = gby ? mval : 0.f;

        const int lbase = di * LDS_COLS + xbase + dj;
#pragma unroll
        for (int t = 0; t < NCT; ++t) {
            const int off = (t * 16 + mrow) * LDS_CH + lbase;
            v2f a;
            a.x = lfeat[off];
            a.y = lfeat[off + 1];
            acc[t] = __builtin_amdgcn_wmma_f32_16x16x4_f32(
                false, a, false, b, (short)0, acc[t], false, false);
        }
    }

    // ---- repack through LDS so every global store is a full 128B row ----
    __syncthreads();                           // done reading feature tile
    const int owl = 2 * (wave * 4 + p) + s2;   // block-local output column 0..31
#pragma unroll
    for (int t = 0; t < NCT; ++t) {
#pragma unroll
        for (int v = 0; v < 8; ++v) {
            const int c = t * 16 + v + half * 8;
            lfeat[c * OST + s1 * 32 + owl] = acc[t][v];
        }
    }
    __syncthreads();

    // 256 output rows (c, s1); each wave writes 64 of them, 128B coalesced.
    float* out_n = out + (size_t)n * CCH * OHH * OWW;
    for (int i = 0; i < 64; ++i) {
        const int rid = (i << 2) | wave;       // 0..255
        const int c   = rid >> 1;
        const int sl  = rid & 1;
        const float val = lfeat[c * OST + sl * 32 + lane];
        out_n[(size_t)c * (OHH * OWW) + (size_t)(2 * hi + sl) * OWW
              + (wblk * 2 + lane)] = val;
    }
}

extern "C" void kernel_launch(void* const* d_in, const int* in_sizes, int n_in,
                              void* d_out, int out_size, void* d_ws, size_t ws_size,
                              hipStream_t stream) {
    const float* feats = (const float*)d_in[0];  // (2,128,128,128) fp32
    const float* masks = (const float*)d_in[1];  // (2,25,256,256) fp32
    float* out = (float*)d_out;                  // (2,128,256,256) fp32

    dim3 grid(2 * 128 * 8);  // n * hi * (W/16)
    dim3 block(128);         // 4 waves, each owning 4 low-res pixels
    hipLaunchKernelGGL(carafe_wmma_f32_kernel, grid, block, 0, stream,
                       feats, masks, out);
}